// shape_prior_loss_65652870087193
// MI455X (gfx1250) — compile-verified
//
#include <hip/hip_runtime.h>
#include <math.h>

typedef __attribute__((ext_vector_type(2))) float v2f;
typedef __attribute__((ext_vector_type(8))) float v8f;

#define BS   32
#define NP   1024
#define NV   1024
#define NM   1024
#define NROT 30

// ---------------------------------------------------------------- reductions
__device__ __forceinline__ float blockReduceSum256(float v) {
#pragma unroll
  for (int off = 16; off > 0; off >>= 1) v += __shfl_xor(v, off);
  __shared__ float tmp[8];
  int w = threadIdx.x >> 5;
  if ((threadIdx.x & 31) == 0) tmp[w] = v;
  __syncthreads();
  float r = (threadIdx.x < 8) ? tmp[threadIdx.x] : 0.f;
#pragma unroll
  for (int off = 4; off > 0; off >>= 1) r += __shfl_xor(r, off);
  return r;  // valid in thread 0
}

__device__ __forceinline__ float huber3(float d) {
  d = fabsf(d);
  // THRESHOLD = 0.1 : d > t ? d - t/2 : d*d/(2t)
  return d > 0.1f ? d - 0.05f : d * d * 5.0f;
}

// ---------------------------------------------------------------- init
__global__ void init_acc_kernel(float* acc) {
  if (threadIdx.x < 8) acc[threadIdx.x] = 0.f;
}

// -------------------------------------------- inst_shape = prior + deltas,
// deform_loss partial = sum ||deltas||
__global__ __launch_bounds__(256) void inst_deform_kernel(
    const float* __restrict__ deltas, const float* __restrict__ prior,
    float* __restrict__ inst, float* __restrict__ deform_acc) {
  int idx = blockIdx.x * 256 + threadIdx.x;  // 0 .. BS*NV-1
  float nrm = 0.f;
  if (idx < BS * NV) {
    size_t o = (size_t)idx * 3;
    float dx = deltas[o], dy = deltas[o + 1], dz = deltas[o + 2];
    inst[o]     = prior[o]     + dx;
    inst[o + 1] = prior[o + 1] + dy;
    inst[o + 2] = prior[o + 2] + dz;
    nrm = sqrtf(dx * dx + dy * dy + dz * dz);
  }
  float s = blockReduceSum256(nrm);
  if (threadIdx.x == 0) atomicAdd(deform_acc, s);
}

// -------------------------------------------- fused softmax + coords + entropy
// One wave owns one 1024-wide row of assign_mat (32 f32 per lane).
// Single pass over the 128 MB tensor: this is the bandwidth-bound kernel.
__global__ __launch_bounds__(256) void softmax_coords_entropy_kernel(
    const float* __restrict__ assign, const float* __restrict__ inst,
    float* __restrict__ coords, float* __restrict__ ent_acc) {
  __shared__ float ix[NV], iy[NV], iz[NV];
  int b = blockIdx.y;
  for (int v = threadIdx.x; v < NV; v += 256) {
    size_t o = ((size_t)b * NV + v) * 3;
    ix[v] = inst[o]; iy[v] = inst[o + 1]; iz[v] = inst[o + 2];
  }
  __syncthreads();

  int lane = threadIdx.x & 31, wid = threadIdx.x >> 5;
  float entsum = 0.f;
  for (int i = 0; i < 16; i++) {
    int p = blockIdx.x * 128 + wid * 16 + i;
    const float* row = assign + ((size_t)b * NP + p) * NV;
    float x[32];
#pragma unroll
    for (int j = 0; j < 32; j++) x[j] = row[j * 32 + lane];
    float m = x[0];
#pragma unroll
    for (int j = 1; j < 32; j++) m = fmaxf(m, x[j]);
#pragma unroll
    for (int off = 1; off < 32; off <<= 1) m = fmaxf(m, __shfl_xor(m, off));
    float Z = 0.f, cx = 0.f, cy = 0.f, cz = 0.f, px = 0.f;
#pragma unroll
    for (int j = 0; j < 32; j++) {
      int v = j * 32 + lane;
      float e = __expf(x[j] - m);
      Z += e; px += e * x[j];
      cx += e * ix[v]; cy += e * iy[v]; cz += e * iz[v];
    }
#pragma unroll
    for (int off = 1; off < 32; off <<= 1) {
      Z  += __shfl_xor(Z, off);  cx += __shfl_xor(cx, off);
      cy += __shfl_xor(cy, off); cz += __shfl_xor(cz, off);
      px += __shfl_xor(px, off);
    }
    if (lane == 0) {
      float invZ = 1.f / Z;
      size_t o = ((size_t)b * NP + p) * 3;
      coords[o] = cx * invZ; coords[o + 1] = cy * invZ; coords[o + 2] = cz * invZ;
      // -sum p log p = max + logZ - (sum e*x)/Z
      entsum += m + __logf(Z) - px * invZ;
    }
  }
  if (lane == 0) atomicAdd(ent_acc, entsum);
}

// -------------------------------------------- rotation + Huber: loss_br[b][r]
__global__ __launch_bounds__(256) void corr_rot_kernel(
    const float* __restrict__ coords, const float* __restrict__ nocs,
    const unsigned char* __restrict__ mask, float* __restrict__ loss_br) {
  int rI = blockIdx.x;  // 0..29
  int b  = blockIdx.y;  // 0..31
  float theta = (6.283185307179586f / (float)NROT) * (float)rI;
  float ct = __cosf(theta), st = __sinf(theta);
  float sum = 0.f;
  for (int p = threadIdx.x; p < NP; p += 256) {
    if (mask[b * NP + p]) {  // masked-out points: both sides 0 -> huber 0
      size_t o = ((size_t)b * NP + p) * 3;
      float cx = coords[o], cy = coords[o + 1], cz = coords[o + 2];
      float nx = nocs[o],   ny = nocs[o + 1],   nz = nocs[o + 2];
      float rx = nx * ct - nz * st;
      float rz = nx * st + nz * ct;
      sum += huber3(cx - rx) + huber3(cy - ny) + huber3(cz - rz);
    }
  }
  float s = blockReduceSum256(sum);
  if (threadIdx.x == 0) loss_br[b * NROT + rI] = s / (float)NP;
}

// -------------------------------------------- chamfer row-mins via WMMA f32 K=4
// A = {x1, y1, z1, 1}, B = {-2x2, -2y2, -2z2, |x2|^2}  =>  D = |x2|^2 - 2 x1.x2
// so d(M,N) = |x1_M|^2 + D(M,N).  B fragments are pre-packed in LDS as float2
// pairs (one unconditional ds_load_b64 per lane per tile, no exec masking).
// pass 0: rowmin over model for each inst point   (cd term 1)
// pass 1: rowmin over inst  for each model point  (cd term 2)
// pass 2: rowmin over target for each inst point  (sym term 1)
// pass 3: rowmin over inst  for each target point (sym term 2)
__global__ __launch_bounds__(256) void chamfer_wmma_kernel(
    const float* __restrict__ inst, const float* __restrict__ model,
    const long long* __restrict__ sym, float* __restrict__ cham_acc) {
  int b = blockIdx.x, pass = blockIdx.y;
  // region0: {-2x,-2y} pairs (2*NM dwords) | 32-dword pad | region1: {-2z,nsq}
  __shared__ float bpack[2 * NM + 32 + 2 * NM];

  long long s0 = sym[b * 4], s1 = sym[b * 4 + 1], s2 = sym[b * 4 + 2], s3 = sym[b * 4 + 3];
  bool c1 = (s0 == 1) && ((s1 + s2 + s3) > 0);
  bool c2 = (s0 == 0) && (s1 == 1);
  float sgx = c1 ? -1.f : 1.f;
  float sgz = (c1 || c2) ? -1.f : 1.f;

  const float* instB  = inst  + (size_t)b * NV * 3;
  const float* modelB = model + (size_t)b * NM * 3;

  const float *X1, *X2;
  float x1sx = 1.f, x1sz = 1.f, x2sx = 1.f, x2sz = 1.f;
  if (pass == 0)      { X1 = instB;  X2 = modelB; }
  else if (pass == 1) { X1 = modelB; X2 = instB;  }
  else if (pass == 2) { X1 = instB;  X2 = instB;  x2sx = sgx; x2sz = sgz; }
  else                { X1 = instB;  X2 = instB;  x1sx = sgx; x1sz = sgz; }

  for (int v = threadIdx.x; v < NM; v += 256) {
    float x = X2[v * 3] * x2sx, y = X2[v * 3 + 1], z = X2[v * 3 + 2] * x2sz;
    float nsq = x * x + y * y + z * z;
    v2f lo, hi;
    lo[0] = -2.f * x; lo[1] = -2.f * y;
    hi[0] = -2.f * z; hi[1] = nsq;
    *(v2f*)&bpack[2 * v]               = lo;
    *(v2f*)&bpack[2 * NM + 32 + 2 * v] = hi;
  }
  __syncthreads();

  int lane = threadIdx.x & 31, wid = threadIdx.x >> 5;
  int half = lane >> 4;       // 0: supplies K0,K1 ; 1: supplies K2,K3
  int l16  = lane & 15;
  int bbase = half * (2 * NM + 32);  // per-lane region base (dwords)

  float wsum = 0.f;
  for (int strip = wid; strip < NV / 16; strip += 8) {
    int r = strip * 16 + l16;
    float x = X1[r * 3] * x1sx, y = X1[r * 3 + 1], z = X1[r * 3 + 2] * x1sz;
    float nsq = x * x + y * y + z * z;
    // A fragment 16x4: ISA 32-bit A layout; K3 = 1 feeds the |x2|^2 row of B
    v2f afrag;
    afrag[0] = half ? z : x;
    afrag[1] = half ? 1.f : y;
    // row norms for the 8 D rows this lane sees (M = k (+8 for upper half))
    float n1k[8];
#pragma unroll
    for (int k = 0; k < 8; k++) n1k[k] = __shfl(nsq, k + half * 8);
    float acc[8];
#pragma unroll
    for (int k = 0; k < 8; k++) acc[k] = 3.4e38f;

    for (int t = 0; t < NM / 16; t++) {
      int c = t * 16 + l16;
      v2f bfrag = *(const v2f*)&bpack[bbase + 2 * c];
      v8f czero = {};
      v8f dot = __builtin_amdgcn_wmma_f32_16x16x4_f32(
          false, afrag, false, bfrag, (short)0, czero, false, false);
#pragma unroll
      for (int k = 0; k < 8; k++)
        acc[k] = fminf(acc[k], n1k[k] + dot[k]);
    }
    // reduce min over the 16 columns held by this lane-group
#pragma unroll
    for (int k = 0; k < 8; k++) {
#pragma unroll
      for (int off = 1; off < 16; off <<= 1)
        acc[k] = fminf(acc[k], __shfl_xor(acc[k], off));
    }
    if (l16 == 0) {  // lanes 0 (rows 0..7) and 16 (rows 8..15)
      float s = 0.f;
#pragma unroll
      for (int k = 0; k < 8; k++) s += acc[k];
      wsum += s;
    }
  }
#pragma unroll
  for (int off = 16; off > 0; off >>= 1) wsum += __shfl_xor(wsum, off);
  if (lane == 0) atomicAdd(&cham_acc[pass], wsum);
}

// -------------------------------------------- finalize 5 scalars
__global__ void finalize_kernel(const float* __restrict__ acc,
                                const float* __restrict__ loss_br,
                                const long long* __restrict__ sym,
                                float* __restrict__ out) {
  int lane = threadIdx.x;
  float cb = 0.f;
  if (lane < BS) {
    float id = loss_br[lane * NROT];
    float mn = id;
    for (int r = 1; r < NROT; r++) mn = fminf(mn, loss_br[lane * NROT + r]);
    cb = (sym[lane * 4] == 1) ? mn : id;
  }
#pragma unroll
  for (int off = 16; off > 0; off >>= 1) cb += __shfl_xor(cb, off);
  if (lane == 0) {
    const float invBN = 1.f / (float)(BS * NP);
    out[0] = 1.0f   * (cb / (float)BS);                    // corr_loss
    out[1] = 1e-4f  * (acc[0] * invBN);                    // entropy_loss
    out[2] = 5.0f   * ((acc[2] + acc[3]) * invBN);         // cd_loss
    out[3] = 0.01f  * (acc[1] / (float)(BS * NV));         // deform_loss
    out[4] = 1.0f   * (((acc[4] + acc[5]) * invBN) / BS);  // sym_loss
  }
}

// ---------------------------------------------------------------- launch
extern "C" void kernel_launch(void* const* d_in, const int* in_sizes, int n_in,
                              void* d_out, int out_size, void* d_ws, size_t ws_size,
                              hipStream_t stream) {
  const float* assign = (const float*)d_in[0];
  const float* deltas = (const float*)d_in[1];
  const float* prior  = (const float*)d_in[2];
  const float* nocs   = (const float*)d_in[3];
  const float* model  = (const float*)d_in[4];
  const unsigned char* mask = (const unsigned char*)d_in[5];
  const long long* sym = (const long long*)d_in[6];
  float* out = (float*)d_out;
  float* ws  = (float*)d_ws;

  float* acc    = ws;              // [0]=entropy [1]=deform [2..5]=chamfer passes
  float* lossbr = ws + 8;          // 32*30
  float* inst   = ws + 1024;       // 32*1024*3
  float* coords = ws + 1024 + BS * NV * 3;

  init_acc_kernel<<<1, 32, 0, stream>>>(acc);
  inst_deform_kernel<<<(BS * NV + 255) / 256, 256, 0, stream>>>(deltas, prior, inst, acc + 1);
  softmax_coords_entropy_kernel<<<dim3(NP / 128, BS), 256, 0, stream>>>(assign, inst, coords, acc);
  corr_rot_kernel<<<dim3(NROT, BS), 256, 0, stream>>>(coords, nocs, mask, lossbr);
  chamfer_wmma_kernel<<<dim3(BS, 4), 256, 0, stream>>>(inst, model, sym, acc + 2);
  finalize_kernel<<<1, 32, 0, stream>>>(acc, lossbr, sym, out);
}